// Local_Attention_2551210574591
// MI455X (gfx1250) — compile-verified
//
#include <hip/hip_runtime.h>

// ---------------------------------------------------------------------------
// Gaussian local attention, MI455X (gfx1250, wave32).
//   v    = x @ Wg                [8192,512] @ [512,1024]
//   attn = corr(band) @ v        per (b,h): banded [N,N] @ [N,128]
//   out  = attn @ Wout           [8192,1024] @ [1024,512]
// fp32 via V_WMMA_F32_16X16X4_F32. Band half-width 64 (exact to fp32 for
// sigma<=2.5); inv_norm cancels in the row normalization.
// 32x128 per-wave tiles (16 v8f accumulators); all B operands stored in
// pair-interleaved layout so each B fragment is one global_load_b64.
// ---------------------------------------------------------------------------

typedef float v2f __attribute__((ext_vector_type(2)));
typedef float v8f __attribute__((ext_vector_type(8)));

#define SEQ    2048
#define DD     512
#define HH     8
#define DHH    128
#define INNER2 1024
#define BN     8192     // B * SEQ
#define WIN    64

__device__ __forceinline__ v8f wmma4(v2f a, v2f b, v8f c) {
  return __builtin_amdgcn_wmma_f32_16x16x4_f32(
      false, a, false, b, (short)0, c, false, false);
}

__device__ __forceinline__ v8f zero8() {
  v8f z = {0.f, 0.f, 0.f, 0.f, 0.f, 0.f, 0.f, 0.f};
  return z;
}

// ---------------------------------------------------------------------------
// Pack a row-major [rows,cols] matrix into pair-interleaved layout:
//   WI[((k>>1)*cols + c)*2 + (k&1)] = W[k*cols + c]
// so a (B[k],B[k+1]) fragment pair is one 8-byte load.
// ---------------------------------------------------------------------------
__global__ __launch_bounds__(256) void pack_pairs(const float* __restrict__ W,
                                                  float* __restrict__ WI,
                                                  int rows, int cols) {
  const int idx = blockIdx.x * blockDim.x + threadIdx.x;
  if (idx >= rows * cols) return;
  const int k = idx / cols, c = idx % cols;
  WI[((size_t)(k >> 1) * cols + c) * 2 + (k & 1)] = W[idx];
}

// ---------------------------------------------------------------------------
// Kernel 1: V projection. One wave -> 32 rows x 128 cols of x@Wg, stored
// pair-interleaved per (b,h) into VI for b64 B-fragment reads in kernel 2.
// ---------------------------------------------------------------------------
__global__ __launch_bounds__(256) void vproj_kernel(const float* __restrict__ X,
                                                    const float* __restrict__ WgI,
                                                    float* __restrict__ VI) {
  const int lane = threadIdx.x & 31;
  const int wid  = (blockIdx.x * blockDim.x + threadIdx.x) >> 5;
  const int rowTile = wid >> 3;           // 0..255  (32-row tiles)
  const int colBlk  = wid & 7;            // 0..7    (128-col blocks)
  const int row0 = rowTile * 32;
  const int col0 = colBlk * 128;
  const int mr = lane & 15, half = lane >> 4;

  v8f acc[2][8];
#pragma unroll
  for (int s = 0; s < 2; ++s)
#pragma unroll
    for (int t = 0; t < 8; ++t) acc[s][t] = zero8();

  const float* xr0 = X + (size_t)(row0 + mr) * DD + 2 * half;
  const float* xr1 = xr0 + 16 * DD;

#pragma unroll 4
  for (int k = 0; k < DD; k += 4) {
    const v2f a0 = *(const v2f*)(xr0 + k);
    const v2f a1 = *(const v2f*)(xr1 + k);
    const v2f* wp = (const v2f*)WgI +
                    (size_t)((k >> 1) + half) * INNER2 + col0 + mr;
#pragma unroll
    for (int t = 0; t < 8; ++t) {
      const v2f bf = wp[t * 16];
      acc[0][t] = wmma4(a0, bf, acc[0][t]);
      acc[1][t] = wmma4(a1, bf, acc[1][t]);
    }
  }

#pragma unroll
  for (int s = 0; s < 2; ++s)
#pragma unroll
    for (int t = 0; t < 8; ++t) {
      const int gc = col0 + t * 16 + mr;
      const int h = gc >> 7, dh = gc & 127;
#pragma unroll
      for (int i = 0; i < 4; ++i) {
        const int gr = row0 + 16 * s + 8 * half + 2 * i;   // even
        const int b = gr >> 11, n = gr & 2047;
        v2f o;
        o.x = acc[s][t][2 * i];
        o.y = acc[s][t][2 * i + 1];
        *(v2f*)(VI + (size_t)(b * HH + h) * SEQ * DHH +
                ((size_t)(n >> 1) * DHH + dh) * 2) = o;
      }
    }
}

// ---------------------------------------------------------------------------
// Kernel 2: banded Gaussian attention. One wave -> one (b,h) x 32-row tile,
// full DH=128. corr weights generated in registers (v_exp_f32); rows
// rescaled by banded row sums after accumulation.
// ---------------------------------------------------------------------------
__global__ __launch_bounds__(256) void attn_kernel(const float* __restrict__ VI,
                                                   const float* __restrict__ sigma,
                                                   float* __restrict__ ATT) {
  const int lane = threadIdx.x & 31;
  const int wid  = (blockIdx.x * blockDim.x + threadIdx.x) >> 5;
  const int bh = wid >> 6;                // 0..31
  const int rt = wid & 63;                // 0..63  (32-row tiles)
  const int b = bh >> 3, h = bh & 7;
  const int n0 = rt * 32;
  const int mr = lane & 15, half = lane >> 4;

  const float sg = sigma[h];
  const float inv2s2 = 1.0f / (2.0f * sg * sg);

  const v2f* vb = (const v2f*)(VI + (size_t)bh * SEQ * DHH);

  int m_lo = n0 - WIN;      if (m_lo < 0)   m_lo = 0;
  int m_hi = n0 + 32 + WIN; if (m_hi > SEQ) m_hi = SEQ;

  v8f acc[2][8];
#pragma unroll
  for (int s = 0; s < 2; ++s)
#pragma unroll
    for (int t = 0; t < 8; ++t) acc[s][t] = zero8();
  float rp0 = 0.0f, rp1 = 0.0f;

  for (int m0 = m_lo; m0 < m_hi; m0 += 4) {
    const int mA = m0 + 2 * half;                 // even; lane K cols mA, mA+1
    const float d0a = (float)(n0 + mr - mA);
    const float d0b = d0a - 1.0f;
    const float d1a = d0a + 16.0f;
    const float d1b = d1a - 1.0f;
    const float w0a = __expf(-d0a * d0a * inv2s2);
    const float w0b = __expf(-d0b * d0b * inv2s2);
    const float w1a = __expf(-d1a * d1a * inv2s2);
    const float w1b = __expf(-d1b * d1b * inv2s2);
    rp0 += w0a + w0b;
    rp1 += w1a + w1b;
    v2f a0, a1;
    a0.x = w0a; a0.y = w0b;
    a1.x = w1a; a1.y = w1b;
    const v2f* vp = vb + (size_t)(mA >> 1) * DHH + mr;
#pragma unroll
    for (int t = 0; t < 8; ++t) {
      const v2f bf = vp[t * 16];
      acc[0][t] = wmma4(a0, bf, acc[0][t]);
      acc[1][t] = wmma4(a1, bf, acc[1][t]);
    }
  }

  const float rs0 = rp0 + __shfl_xor(rp0, 16, 32);
  const float rs1 = rp1 + __shfl_xor(rp1, 16, 32);
  float inv0[8], inv1[8];
#pragma unroll
  for (int r = 0; r < 8; ++r) {
    inv0[r] = 1.0f / __shfl(rs0, r + 8 * half, 32);
    inv1[r] = 1.0f / __shfl(rs1, r + 8 * half, 32);
  }

#pragma unroll
  for (int t = 0; t < 8; ++t)
#pragma unroll
    for (int r = 0; r < 8; ++r) {
      const int n_a = n0 + r + 8 * half;
      const int n_b = n_a + 16;
      const size_t col = (size_t)h * DHH + t * 16 + mr;
      ATT[((size_t)(b * SEQ + n_a)) * INNER2 + col] = acc[0][t][r] * inv0[r];
      ATT[((size_t)(b * SEQ + n_b)) * INNER2 + col] = acc[1][t][r] * inv1[r];
    }
}

// ---------------------------------------------------------------------------
// Kernel 3: output projection. attn[8192,1024] @ Wout[1024,512] -> out.
// ---------------------------------------------------------------------------
__global__ __launch_bounds__(256) void oproj_kernel(const float* __restrict__ ATT,
                                                    const float* __restrict__ WoutI,
                                                    float* __restrict__ OUT) {
  const int lane = threadIdx.x & 31;
  const int wid  = (blockIdx.x * blockDim.x + threadIdx.x) >> 5;
  const int rowTile = wid >> 2;           // 0..255 (32-row tiles)
  const int colBlk  = wid & 3;            // 0..3   (128-col blocks)
  const int row0 = rowTile * 32;
  const int col0 = colBlk * 128;
  const int mr = lane & 15, half = lane >> 4;

  v8f acc[2][8];
#pragma unroll
  for (int s = 0; s < 2; ++s)
#pragma unroll
    for (int t = 0; t < 8; ++t) acc[s][t] = zero8();

  const float* ar0 = ATT + (size_t)(row0 + mr) * INNER2 + 2 * half;
  const float* ar1 = ar0 + 16 * INNER2;

#pragma unroll 4
  for (int k = 0; k < INNER2; k += 4) {
    const v2f a0 = *(const v2f*)(ar0 + k);
    const v2f a1 = *(const v2f*)(ar1 + k);
    const v2f* wp = (const v2f*)WoutI +
                    (size_t)((k >> 1) + half) * DD + col0 + mr;
#pragma unroll
    for (int t = 0; t < 8; ++t) {
      const v2f bf = wp[t * 16];
      acc[0][t] = wmma4(a0, bf, acc[0][t]);
      acc[1][t] = wmma4(a1, bf, acc[1][t]);
    }
  }

#pragma unroll
  for (int s = 0; s < 2; ++s)
#pragma unroll
    for (int t = 0; t < 8; ++t)
#pragma unroll
      for (int r = 0; r < 8; ++r) {
        const int gr = row0 + 16 * s + r + 8 * half;
        OUT[(size_t)gr * DD + col0 + t * 16 + mr] = acc[s][t][r];
      }
}

// ---------------------------------------------------------------------------
extern "C" void kernel_launch(void* const* d_in, const int* in_sizes, int n_in,
                              void* d_out, int out_size, void* d_ws, size_t ws_size,
                              hipStream_t stream) {
  const float* X     = (const float*)d_in[0];   // [4,2048,512]
  const float* Wg    = (const float*)d_in[1];   // [512,1024]
  const float* Wout  = (const float*)d_in[2];   // [1024,512]
  const float* sigma = (const float*)d_in[3];   // [8]
  float* OUT = (float*)d_out;                   // [4,2048,512]

  float* VI    = (float*)d_ws;                          // 8.39M floats
  float* ATT   = VI    + (size_t)BN * INNER2;           // 8.39M floats
  float* WgI   = ATT   + (size_t)BN * INNER2;           // 0.52M floats
  float* WoutI = WgI   + (size_t)DD * INNER2;           // 0.52M floats

  pack_pairs<<<dim3((DD * INNER2) / 256), dim3(256), 0, stream>>>(Wg, WgI, DD, INNER2);
  pack_pairs<<<dim3((DD * INNER2) / 256), dim3(256), 0, stream>>>(Wout, WoutI, INNER2, DD);

  vproj_kernel<<<dim3(256), dim3(256), 0, stream>>>(X, WgI, VI);      // 2048 waves
  attn_kernel <<<dim3(256), dim3(256), 0, stream>>>(VI, sigma, ATT);  // 2048 waves
  oproj_kernel<<<dim3(128), dim3(256), 0, stream>>>(ATT, WoutI, OUT); // 1024 waves
}